// LSTMFused_80187039416532
// MI455X (gfx1250) — compile-verified
//
#include <hip/hip_runtime.h>
#include <hip/hip_bf16.h>

// ---------------------------------------------------------------------------
// Fused 2-layer block-diagonal multi-head LSTM for MI455X (gfx1250, wave32).
// All matrix math runs on V_WMMA_F32_16X16X4_F32 (exact fp32, matches the
// fp32 reference; no precision downcast since the scan compounds error
// over 1024 steps).
// ---------------------------------------------------------------------------

typedef __attribute__((ext_vector_type(2))) float v2f;
typedef __attribute__((ext_vector_type(8))) float v8f;

__device__ __forceinline__ v8f wmma_f32_16x16x4(v2f a, v2f b, v8f c) {
  // 8 args: (neg_a, A, neg_b, B, c_mod, C, reuse_a, reuse_b)
  return __builtin_amdgcn_wmma_f32_16x16x4_f32(false, a, false, b,
                                               (short)0, c, false, false);
}

__device__ __forceinline__ float sigmoidf_(float x) {
  return 1.0f / (1.0f + __expf(-x));
}

constexpr int B_  = 32;     // batch
constexpr int T_  = 1024;   // sequence length
constexpr int IN_ = 1024;   // input size
constexpr int NH_ = 8;      // heads
constexpr int D_  = 128;    // head dim
constexpr int H_  = 1024;   // NH*D
constexpr int G_  = 4;      // gates (i,f,z,o)
constexpr int ND_ = 512;    // G*D (per-head gate width, d-major: idx = d*4+g)
constexpr int L_  = 2;      // layers

// workspace layout (in floats); total ~675 MB
constexpr size_t WX_N   = (size_t)T_ * B_ * NH_ * ND_;   // 134217728
constexpr size_t HS_N   = (size_t)T_ * B_ * NH_ * D_;    //  33554432
constexpr size_t WX_OFF = 0;
constexpr size_t HS_OFF = WX_N;
constexpr size_t RP_OFF = WX_N + HS_N;
constexpr size_t RP_PER_HEAD = 32 * 32 * 32 * 2;         // 65536 floats

// ---------------------------------------------------------------------------
// Pack R (L,NH,D,G,D) into per-lane WMMA B-fragment order so the scan's
// recurrent GEMM does one coalesced 8B/lane load per fragment (L2-resident).
// Layout: [layer][head][ntile(32)][ktile(32)][lane(32)][2 floats]
//   B[k][N] = R[layer][head][d][g][j], j=k, N=d*4+g
//   lane l, reg v -> K = kt*4 + (l/16)*2 + v ; N = nt*16 + (l%16)
// ---------------------------------------------------------------------------
__global__ __launch_bounds__(256) void packR_kernel(
    const float* __restrict__ R, float* __restrict__ Rp) {
  int idx  = blockIdx.x * 256 + threadIdx.x;   // [0, 524288)
  int lane = idx & 31;
  int kt   = (idx >> 5) & 31;
  int nt   = (idx >> 10) & 31;
  int hd   = (idx >> 15) & 7;
  int ly   = idx >> 18;
  int k0 = kt * 4 + ((lane >> 4) << 1);
  int N  = nt * 16 + (lane & 15);
  int d  = N >> 2, g = N & 3;
  const float* src = R + ((((size_t)ly * NH_ + hd) * D_ + d) * G_ + g) * D_;
  float2 v;
  v.x = src[k0];
  v.y = src[k0 + 1];
  *(float2*)(Rp + (size_t)idx * 2) = v;
}

// ---------------------------------------------------------------------------
// Input projection: Wx[t][b][n][d][g] = sum_k x[b][t][k]*W[g*1024+n*128+d][k] + Wb
// Each wave: 16 rows (m = b*T+t) x (16 d's x 4 gates) via 4 WMMA accumulators,
// sharing one A fragment per k-chunk. Output staged through LDS so global
// stores are contiguous float4 (256B per row of the Wx layout).
// ---------------------------------------------------------------------------
__global__ __launch_bounds__(256) void gemmX_kernel(
    const float* __restrict__ x, const float* __restrict__ W,
    const float* __restrict__ Wb, float* __restrict__ Wx) {
  __shared__ float stage[8][16][68];           // 68: float4-aligned pad
  int tid = threadIdx.x, wv = tid >> 5, lane = tid & 31;
  int wgid = blockIdx.x * 8 + wv;              // [0, 131072)
  int rowTile  = wgid >> 6;                    // 2048 row tiles
  int colGroup = wgid & 63;                    // 8 heads x 8 d-tiles
  int n  = colGroup >> 3;
  int d0 = (colGroup & 7) << 4;
  int laneM = lane & 15;
  int ksel  = (lane >> 4) << 1;
  int rowBase = rowTile << 4;

  const float* aPtr = x + (size_t)(rowBase + laneM) * IN_ + ksel;
  const float* bPtr = W + (size_t)(n * D_ + d0 + laneM) * IN_ + ksel;

  v8f acc[4] = {};
  for (int k0 = 0; k0 < IN_; k0 += 4) {
    v2f a = *(const v2f*)(aPtr + k0);
#pragma unroll
    for (int c = 0; c < 4; ++c) {              // c = gate
      v2f b = *(const v2f*)(bPtr + (size_t)c * H_ * IN_ + k0);
      acc[c] = wmma_f32_16x16x4(a, b, acc[c]);
    }
  }

  int rsel = (lane >> 4) << 3;
#pragma unroll
  for (int c = 0; c < 4; ++c) {
    float bias = Wb[c * H_ + n * D_ + d0 + laneM];
#pragma unroll
    for (int v = 0; v < 8; ++v)
      stage[wv][rsel + v][laneM * 4 + c] = acc[c][v] + bias;   // [d][g]
  }
  __syncthreads();
#pragma unroll
  for (int it = 0; it < 8; ++it) {
    int r = it * 2 + (lane >> 4);
    int m = rowBase + r;
    int bb = m >> 10, tt = m & 1023;           // m = b*T + t
    float4 val = *(const float4*)&stage[wv][r][laneM * 4];
    *(float4*)(Wx + ((size_t)((size_t)tt * B_ + bb) * NH_ + n) * ND_ +
               (size_t)d0 * 4 + laneM * 4) = val;
  }
}

// ---------------------------------------------------------------------------
// Inter-layer projection: out[r][nn] = sum_j hs[r][j]*W2[nn][j] + W2b[nn]
// r = (t*B+b)*NH+n (262144 rows), nn = d*4+g (512) -> row-major into Wx.
// ---------------------------------------------------------------------------
__global__ __launch_bounds__(256) void gemmH_kernel(
    const float* __restrict__ hs, const float* __restrict__ W2,
    const float* __restrict__ W2b, float* __restrict__ Wx) {
  __shared__ float stage[8][16][68];
  int tid = threadIdx.x, wv = tid >> 5, lane = tid & 31;
  int wgid = blockIdx.x * 8 + wv;              // [0, 131072)
  int rowTile = wgid >> 3;                     // 16384 row tiles
  int cg = wgid & 7;
  int laneM = lane & 15;
  int ksel  = (lane >> 4) << 1;
  int rowBase = rowTile << 4;
  int nnB = cg * 64;

  const float* aPtr = hs + (size_t)(rowBase + laneM) * D_ + ksel;
  const float* bPtr = W2 + (size_t)(nnB + laneM) * D_ + ksel;

  v8f acc[4] = {};
  for (int k0 = 0; k0 < D_; k0 += 4) {
    v2f a = *(const v2f*)(aPtr + k0);
#pragma unroll
    for (int c = 0; c < 4; ++c) {
      v2f b = *(const v2f*)(bPtr + (size_t)(c * 16) * D_ + k0);
      acc[c] = wmma_f32_16x16x4(a, b, acc[c]);
    }
  }

  int rsel = (lane >> 4) << 3;
#pragma unroll
  for (int c = 0; c < 4; ++c) {
    float bias = W2b[nnB + c * 16 + laneM];
#pragma unroll
    for (int v = 0; v < 8; ++v)
      stage[wv][rsel + v][c * 16 + laneM] = acc[c][v] + bias;
  }
  __syncthreads();
#pragma unroll
  for (int it = 0; it < 8; ++it) {
    int r = it * 2 + (lane >> 4);
    float4 val = *(const float4*)&stage[wv][r][laneM * 4];
    *(float4*)(Wx + (size_t)(rowBase + r) * ND_ + nnB + laneM * 4) = val;
  }
}

// ---------------------------------------------------------------------------
// Recurrent scan. One block per (head, batch-half of 16): per step an
// M=16 x N=512 x K=128 fp32 WMMA GEMM (h in LDS, R fragments streamed from
// the L2-resident pre-packed buffer), then fused gate math with c in regs.
// 512 threads = 16 waves; wave w owns N-tiles 2w, 2w+1.
// ---------------------------------------------------------------------------
#define HSTRIDE 132   // 16 batch rows, padded (bank-conflict-free A frags)
#define PSTRIDE 516   // pre rows padded to keep float4 alignment

__global__ __launch_bounds__(512) void lstmScan_kernel(
    const float* __restrict__ Wx, const float* __restrict__ Rp,
    const float* __restrict__ Bg, float* __restrict__ hOut,
    float* __restrict__ hnOut, float* __restrict__ cnOut, int finalLayer) {
  __shared__ float h_lds[16 * HSTRIDE];
  __shared__ float pre_lds[16 * PSTRIDE];

  int tid = threadIdx.x, wv = tid >> 5, lane = tid & 31;
  int head = blockIdx.x & 7;
  int bgrp = blockIdx.x >> 3;                  // 0 or 1 (batch half)
  int laneM = lane & 15;
  int ksel  = (lane >> 4) << 1;

  for (int i = tid; i < 16 * HSTRIDE; i += 512) h_lds[i] = 0.0f;

  // GEMM-phase addressing
  int aOff = laneM * HSTRIDE + ksel;           // A frag base in h_lds
  const float* rpHead = Rp + (size_t)head * RP_PER_HEAD;
  const float* rpA = rpHead + (size_t)(wv * 2) * 2048 + lane * 2;
  const float* rpB = rpA + 2048;
  int rsel = (lane >> 4) << 3;
  int col0 = wv * 32 + laneM;                  // pre column for acc0

  // Gate-phase mapping: thread owns (b_loc = bl+4i, d) for i=0..3
  int bl = tid >> 7;                           // 0..3
  int d  = tid & 127;
  float creg[4] = {0.f, 0.f, 0.f, 0.f};
  float4 bias = *(const float4*)(Bg + ((size_t)head * D_ + d) * G_);

  for (int t = 0; t < T_; ++t) {
    __syncthreads();                           // h_lds ready / pre_lds free

    // Issue Wx loads early so HBM latency hides under the GEMM.
    float4 wxv[4];
#pragma unroll
    for (int i = 0; i < 4; ++i) {
      int bglob = bgrp * 16 + bl + 4 * i;
      const float* p = Wx +
          ((size_t)((size_t)t * B_ + bglob) * NH_ + head) * ND_ + d * 4;
      wxv[i] = *(const float4*)p;
      if (t + 1 < T_)
        __builtin_prefetch(p + (size_t)B_ * NH_ * ND_, 0, 1);
    }

    // pre = h @ R^T  (two 16x16 output tiles per wave, K=128)
    v8f acc0 = {}, acc1 = {};
    for (int kt = 0; kt < 32; ++kt) {
      v2f a  = *(const v2f*)&h_lds[aOff + kt * 4];
      v2f b0 = *(const v2f*)(rpA + kt * 64);
      v2f b1 = *(const v2f*)(rpB + kt * 64);
      acc0 = wmma_f32_16x16x4(a, b0, acc0);
      acc1 = wmma_f32_16x16x4(a, b1, acc1);
    }
#pragma unroll
    for (int v = 0; v < 8; ++v) {
      pre_lds[(rsel + v) * PSTRIDE + col0]      = acc0[v];
      pre_lds[(rsel + v) * PSTRIDE + col0 + 16] = acc1[v];
    }
    __syncthreads();                           // pre ready

    // gates: pre + wx + bias -> sigmoid/tanh -> c,h update
#pragma unroll
    for (int i = 0; i < 4; ++i) {
      int b_loc = bl + 4 * i;
      int bglob = bgrp * 16 + b_loc;
      float4 pr = *(const float4*)&pre_lds[b_loc * PSTRIDE + d * 4];
      float gi = sigmoidf_(pr.x + wxv[i].x + bias.x);
      float gf = sigmoidf_(pr.y + wxv[i].y + bias.y);
      float gz = tanhf    (pr.z + wxv[i].z + bias.z);
      float go = sigmoidf_(pr.w + wxv[i].w + bias.w);
      float cc = gf * creg[i] + gi * gz;
      creg[i] = cc;
      float hh = go * tanhf(cc);
      h_lds[b_loc * HSTRIDE + d] = hh;
      if (finalLayer) {        // d_out h: (B,T,H)
        hOut[(size_t)bglob * T_ * H_ + (size_t)t * H_ + head * D_ + d] = hh;
      } else {                 // hs scratch: (T,B,NH,D)
        hOut[((size_t)((size_t)t * B_ + bglob) * NH_ + head) * D_ + d] = hh;
      }
      if (t == T_ - 1) {
        hnOut[(size_t)bglob * H_ + head * D_ + d] = hh;
        cnOut[(size_t)bglob * H_ + head * D_ + d] = cc;
      }
    }
  }
}

// ---------------------------------------------------------------------------
extern "C" void kernel_launch(void* const* d_in, const int* in_sizes, int n_in,
                              void* d_out, int out_size, void* d_ws,
                              size_t ws_size, hipStream_t stream) {
  const float* x   = (const float*)d_in[0];
  const float* W   = (const float*)d_in[1];
  const float* Wb  = (const float*)d_in[2];
  const float* W2  = (const float*)d_in[3];
  const float* W2b = (const float*)d_in[4];
  const float* R   = (const float*)d_in[5];
  const float* Bg  = (const float*)d_in[6];
  float* out = (float*)d_out;
  float* ws  = (float*)d_ws;

  float* Wx = ws + WX_OFF;   // (T,B,NH,D,G) gate preactivations  536 MB
  float* hs = ws + HS_OFF;   // (T,B,NH,D)   layer-0 hidden states 134 MB
  float* Rp = ws + RP_OFF;   // packed R fragments                  4 MB

  float* hn = out + (size_t)B_ * T_ * H_;
  float* cn = hn + (size_t)L_ * B_ * H_;

  packR_kernel<<<2048, 256, 0, stream>>>(R, Rp);
  gemmX_kernel<<<16384, 256, 0, stream>>>(x, W, Wb, Wx);
  lstmScan_kernel<<<16, 512, 0, stream>>>(Wx, Rp, Bg, hs, hn, cn, 0);
  gemmH_kernel<<<16384, 256, 0, stream>>>(hs, W2, W2b, Wx);
  lstmScan_kernel<<<16, 512, 0, stream>>>(
      Wx, Rp + (size_t)NH_ * RP_PER_HEAD, Bg + (size_t)NH_ * D_ * G_,
      out, hn + (size_t)B_ * H_, cn + (size_t)B_ * H_, 1);
}